// ParallelSelfAttention_13718125543584
// MI455X (gfx1250) — compile-verified
//
#include <hip/hip_runtime.h>
#include <cmath>

typedef float v2f __attribute__((ext_vector_type(2)));
typedef float v8f __attribute__((ext_vector_type(8)));

constexpr int SQ = 1024, SK = 2048, B = 2, NP = 16, HN = 128, TK = 16;
constexpr int BN      = B * NP;    // 32 flattened batch*heads
constexpr int KPAIRS  = SK / 32;   // 64 iterations, 2 key tiles (32 keys) each
constexpr int KCHUNKS = HN / 4;    // 32 fp32-WMMA K-steps over hn=128

__global__ __launch_bounds__(32)
void knn_attn_fused_kernel(const float* __restrict__ q,
                           const float* __restrict__ k,
                           const float* __restrict__ v,
                           float* __restrict__ out)
{
    // LDS. Pads: qlds stride 130 (bank shift 2/row), sc/tv stride 18 so the
    // two lane-halves (rows j and j+8) land on disjoint bank sets.
    __shared__ float qlds[16][HN + 2];
    __shared__ float sc[2][16][18];   // two score tiles per iteration
    __shared__ float tv[16][18];
    __shared__ int   ti[16][18];
    __shared__ float pr[16][18];

    const int s0   = blockIdx.x * 16;   // first query row of this tile
    const int n    = blockIdx.y;        // flattened (b, head)
    const int lane = threadIdx.x;
    const int colL = lane & 15;
    const int hi   = lane >> 4;

    // ---- stage Q tile [16 x 128] into LDS (512B coalesced per row) ----
    for (int i = 0; i < 16; ++i) {
        const float4 qv = *reinterpret_cast<const float4*>(
            q + ((size_t)(s0 + i) * BN + n) * HN + 4 * lane);
        *reinterpret_cast<float4*>(&qlds[i][4 * lane]) = qv;
    }
    if (lane < 16) {
        #pragma unroll
        for (int u = 0; u < TK; ++u) { tv[lane][u] = -__builtin_inff(); ti[lane][u] = 0; }
    }
    __syncthreads();

    // ---- A fragments for all K-chunks (reused across all key tiles) ----
    // fp32 16x16x4 A layout: lanes 0-15 row=lane K={4k,4k+1}; lanes 16-31 K={4k+2,4k+3}
    v2f a[KCHUNKS];
    #pragma unroll
    for (int kk = 0; kk < KCHUNKS; ++kk) {
        a[kk][0] = qlds[colL][4 * kk + 2 * hi];
        a[kk][1] = qlds[colL][4 * kk + 2 * hi + 1];
    }

    float mn = -__builtin_inff();  // current min of top-16 (valid for lanes<16)
    int   mp = 0;                  // its slot

    // ---- fused scores GEMM + running top-16; 2 key tiles / iteration ----
    for (int t = 0; t < KPAIRS; ++t) {
        // B fragments direct from global: keys 32t+colL and 32t+16+colL.
        // kk loop walks 16B steps so each 128B line is fully consumed.
        const float* krow0 = k + ((size_t)(32 * t + colL)      * BN + n) * HN + 2 * hi;
        const float* krow1 = k + ((size_t)(32 * t + 16 + colL) * BN + n) * HN + 2 * hi;
        v8f c0 = {}, c1 = {};   // two independent accumulation chains (ILP)
        #pragma unroll
        for (int kk = 0; kk < KCHUNKS; ++kk) {
            const v2f b0 = *reinterpret_cast<const v2f*>(krow0 + 4 * kk);
            const v2f b1 = *reinterpret_cast<const v2f*>(krow1 + 4 * kk);
            c0 = __builtin_amdgcn_wmma_f32_16x16x4_f32(
                     false, a[kk], false, b0, (short)0, c0, false, false);
            c1 = __builtin_amdgcn_wmma_f32_16x16x4_f32(
                     false, a[kk], false, b1, (short)0, c1, false, false);
        }
        // spill both 16x16 score tiles: VGPR j -> row j (lanes 0-15) / j+8
        #pragma unroll
        for (int j = 0; j < 8; ++j) {
            sc[0][j + 8 * hi][colL] = c0[j];
            sc[1][j + 8 * hi][colL] = c1[j];
        }
        __syncthreads();

        // lane r maintains row r's top-16 over the 32 fresh candidates
        // (ascending key order preserves lowest-index tie-breaking)
        if (lane < 16) {
            for (int cc = 0; cc < 32; ++cc) {
                const float val = sc[cc >> 4][lane][cc & 15];
                if (val > mn) {
                    tv[lane][mp] = val;
                    ti[lane][mp] = 32 * t + cc;
                    mn = tv[lane][0]; mp = 0;
                    #pragma unroll
                    for (int u = 1; u < TK; ++u) {
                        const float x = tv[lane][u];
                        if (x < mn) { mn = x; mp = u; }
                    }
                }
            }
        }
        __syncthreads();
    }

    // ---- softmax over the 16 kept scores (attn == score/sqrt(hn) exactly) ----
    if (lane < 16) {
        const float inv_norm = 0.08838834764831845f;  // 1/sqrt(128)
        float m = -__builtin_inff();
        #pragma unroll
        for (int u = 0; u < TK; ++u) m = fmaxf(m, tv[lane][u] * inv_norm);
        float e[TK], s = 0.f;
        #pragma unroll
        for (int u = 0; u < TK; ++u) {
            e[u] = __expf(tv[lane][u] * inv_norm - m);
            s += e[u];
        }
        const float r = 1.f / s;
        #pragma unroll
        for (int u = 0; u < TK; ++u) pr[lane][u] = e[u] * r;
    }
    __syncthreads();

    // ---- gather V + weighted sum; each lane owns 4 output channels ----
    for (int rI = 0; rI < 16; ++rI) {
        float4 acc = {0.f, 0.f, 0.f, 0.f};
        #pragma unroll
        for (int u = 0; u < TK; ++u) {
            const float w  = pr[rI][u];            // LDS broadcast
            const int  key = ti[rI][u];
            const float4 vv = *reinterpret_cast<const float4*>(
                v + ((size_t)key * BN + n) * HN + 4 * lane);  // 512B coalesced
            acc.x += w * vv.x; acc.y += w * vv.y;
            acc.z += w * vv.z; acc.w += w * vv.w;
        }
        // out[b][np][s][h] flat = (n*SQ + s)*HN + h
        *reinterpret_cast<float4*>(
            out + ((size_t)n * SQ + (s0 + rI)) * HN + 4 * lane) = acc;
    }
}

extern "C" void kernel_launch(void* const* d_in, const int* in_sizes, int n_in,
                              void* d_out, int out_size, void* d_ws, size_t ws_size,
                              hipStream_t stream) {
    (void)in_sizes; (void)n_in; (void)out_size; (void)d_ws; (void)ws_size;
    const float* q = (const float*)d_in[0];
    const float* k = (const float*)d_in[1];
    const float* v = (const float*)d_in[2];
    // d_in[3] = knn_key_count (==16), baked in as TK at compile time.
    float* out = (float*)d_out;
    dim3 grid(SQ / 16, BN);   // 64 query tiles x 32 (batch*head)
    knn_attn_fused_kernel<<<grid, dim3(32), 0, stream>>>(q, k, v, out);
}